// Word2VecModel_38543036514814
// MI455X (gfx1250) — compile-verified
//
#include <hip/hip_runtime.h>
#include <hip/hip_bf16.h>

#define VOCAB 100000
#define EMB   128
#define HID   300
#define BATCH 8192
#define SEQL  200

typedef __attribute__((ext_vector_type(2))) float v2f;
typedef __attribute__((ext_vector_type(8))) float v8f;

// ---------------------------------------------------------------------------
// Kernel 1: ragged embedding-bag mean pooling (the bandwidth-bound kernel:
// ~840 MB of embedding-row gathers; table is L2-resident at 51.2 MB).
//
// One wave32 per batch row. Each lane owns 4 consecutive floats of the
// 128-float embedding row (float4 -> global_load_b128, 512B coalesced per
// wave per token). Token IDs are loaded cooperatively (32 at a time) and
// broadcast with __shfl (ds_bpermute).
//
// The full-chunk path is a fixed-bound loop unrolled x8 so the compiler can
// clause 8 independent b128 loads per wave -> ~8 outstanding loads for
// memory-level parallelism on the critical path. Dynamic-bound tail loop
// handles len%32.
// ---------------------------------------------------------------------------
__global__ void __launch_bounds__(256)
embbag_mean_kernel(const int* __restrict__ tokens,
                   const int* __restrict__ lengths,
                   const float* __restrict__ table,
                   float* __restrict__ pooled) {
  const int wave = (int)((blockIdx.x * blockDim.x + threadIdx.x) >> 5);
  const int lane = (int)(threadIdx.x & 31);
  if (wave >= BATCH) return;

  const int len = lengths[wave];                  // 1..200 guaranteed
  const int* trow = tokens + (long)wave * SEQL;
  const float* tbl = table + lane * 4;            // lane's 16B slice base

  float4 acc = make_float4(0.f, 0.f, 0.f, 0.f);

  int tb = 0;
  // ---- full chunks of 32 tokens: unrolled x8 for load clustering ----
  for (; tb + 32 <= len; tb += 32) {
    const int myTok = trow[tb + lane];            // coalesced id load
#pragma unroll
    for (int j0 = 0; j0 < 32; j0 += 8) {
      int t0 = __shfl(myTok, j0 + 0, 32);
      int t1 = __shfl(myTok, j0 + 1, 32);
      int t2 = __shfl(myTok, j0 + 2, 32);
      int t3 = __shfl(myTok, j0 + 3, 32);
      int t4 = __shfl(myTok, j0 + 4, 32);
      int t5 = __shfl(myTok, j0 + 5, 32);
      int t6 = __shfl(myTok, j0 + 6, 32);
      int t7 = __shfl(myTok, j0 + 7, 32);
      const float4 v0 = *(const float4*)(tbl + (long)t0 * EMB);
      const float4 v1 = *(const float4*)(tbl + (long)t1 * EMB);
      const float4 v2 = *(const float4*)(tbl + (long)t2 * EMB);
      const float4 v3 = *(const float4*)(tbl + (long)t3 * EMB);
      const float4 v4 = *(const float4*)(tbl + (long)t4 * EMB);
      const float4 v5 = *(const float4*)(tbl + (long)t5 * EMB);
      const float4 v6 = *(const float4*)(tbl + (long)t6 * EMB);
      const float4 v7 = *(const float4*)(tbl + (long)t7 * EMB);
      acc.x += v0.x + v1.x + v2.x + v3.x + v4.x + v5.x + v6.x + v7.x;
      acc.y += v0.y + v1.y + v2.y + v3.y + v4.y + v5.y + v6.y + v7.y;
      acc.z += v0.z + v1.z + v2.z + v3.z + v4.z + v5.z + v6.z + v7.z;
      acc.w += v0.w + v1.w + v2.w + v3.w + v4.w + v5.w + v6.w + v7.w;
    }
  }

  // ---- tail: remaining len % 32 tokens ----
  const int remain = len - tb;
  if (remain > 0) {
    int myTok = 0;
    if (lane < remain) myTok = trow[tb + lane];
    for (int j = 0; j < remain; ++j) {
      const int tok = __shfl(myTok, j, 32);
      const float4 v = *(const float4*)(tbl + (long)tok * EMB);
      acc.x += v.x; acc.y += v.y; acc.z += v.z; acc.w += v.w;
    }
  }

  const float inv = 1.0f / (float)len;
  float4 o = make_float4(acc.x * inv, acc.y * inv, acc.z * inv, acc.w * inv);
  *(float4*)(pooled + (long)wave * EMB + lane * 4) = o;
}

// ---------------------------------------------------------------------------
// Kernel 2: MLP via V_WMMA_F32_16X16X4_F32.
// One wave per 16-row M-tile of pooled [8192,128]. The wave sweeps all 19
// N-tiles of W1 [128,300 -> pad 304], accumulating C via 32 chained WMMA
// steps (K=4). A fragments are N-tile-invariant and stay in VGPRs.
//
// Padding n>=300: B loads use a clamped column index (in-bounds, no mask
// multiply needed -- C[m][n] depends only on B column n, and padded columns
// are annihilated in the reduction by w2v==0). All lane-divergent selects are
// branchless (v_cndmask), so EXEC stays all-ones as WMMA requires.
//
// Fragment layouts (ISA 7.12.2, 32-bit):
//   A 16x4 : lane = (k/2)*16 + m, vgpr = k%2  -> per-lane contiguous float2
//   B 4x16 : lane = (k/2)*16 + n, vgpr = k%2  -> two loads, stride HID
//   C 16x16: vgpr = v, lanes 0-15: M=v, lanes 16-31: M=v+8, N = lane%16
//
// h@W2 is folded in-register: per-lane partials, then a 16-lane butterfly
// __shfl_xor per half-wave (rows m = v + 8*half), lane 0 of each half writes
// the final softmax output (size-1 axis: e/e == 1, NaN-propagating, DCE-proof).
// ---------------------------------------------------------------------------
__global__ void __launch_bounds__(256)
mlp_wmma_kernel(const float* __restrict__ pooled,
                const float* __restrict__ W1,
                const float* __restrict__ b1,
                const float* __restrict__ W2,
                const float* __restrict__ b2,
                float* __restrict__ out) {
  const int wavesPerBlock = (int)(blockDim.x >> 5);
  const int mtile = (int)blockIdx.x * wavesPerBlock + (int)(threadIdx.x >> 5);
  const int lane  = (int)(threadIdx.x & 31);
  const int half  = lane >> 4;   // K-half for A/B, M-half for C/D
  const int lid   = lane & 15;   // m for A, n-within-tile for B/C
  const int m0    = mtile * 16;

  // A-operand base: row (m0+lid), starting at K offset 2*half
  const float* arow = pooled + (long)(m0 + lid) * EMB + 2 * half;

  float red[8] = {0.f, 0.f, 0.f, 0.f, 0.f, 0.f, 0.f, 0.f};

  const int NTILES = (HID + 15) / 16;  // 19
  for (int nt = 0; nt < NTILES; ++nt) {
    const int n = nt * 16 + lid;
    const int nc = n < HID ? n : (HID - 1);       // clamped, in-bounds
    const float nmask = n < HID ? 1.0f : 0.0f;    // kills padded cols at reduce

    v8f c = {};  // zero accumulator
#pragma unroll
    for (int k0 = 0; k0 < EMB; k0 += 4) {
      const v2f a = *(const v2f*)(arow + k0);
      v2f b;
      const int kb = k0 + 2 * half;
      b.x = W1[(long)(kb + 0) * HID + nc];   // duplicate col for n>=300 is
      b.y = W1[(long)(kb + 1) * HID + nc];   // harmless: masked out below
      c = __builtin_amdgcn_wmma_f32_16x16x4_f32(
          /*neg_a=*/false, a, /*neg_b=*/false, b,
          /*c_mod=*/(short)0, c, /*reuse_a=*/false, /*reuse_b=*/false);
    }

    const float bias = b1[nc];
    const float w2v  = W2[nc] * nmask;  // zero annihilates padded columns
#pragma unroll
    for (int v = 0; v < 8; ++v) {
      float h = c[v] + bias;            // row m = v + 8*half, col n
      h = h > 0.0f ? h : 0.0f;          // relu
      red[v] += h * w2v;                // partial of h @ W2 for row m
    }
  }

  // Reduce across the 16 lanes of each half (columns n) -> logit per row.
#pragma unroll
  for (int v = 0; v < 8; ++v) {
    float s = red[v];
    s += __shfl_xor(s, 1, 32);
    s += __shfl_xor(s, 2, 32);
    s += __shfl_xor(s, 4, 32);
    s += __shfl_xor(s, 8, 32);
    red[v] = s;  // every lane in the half now holds the row sum
  }

  if (lid == 0) {
    const float bb = b2[0];
#pragma unroll
    for (int v = 0; v < 8; ++v) {
      const float logit = red[v] + bb;
      // softmax over a size-1 axis, computed literally:
      // max = logit; e = exp(logit - max); out = e / sum(e) = e / e
      const float e = __expf(logit - logit);
      out[m0 + v + 8 * half] = e / e;
    }
  }
}

// ---------------------------------------------------------------------------
// Launch. Inputs (setup_inputs order): tokens[i32], lengths[i32],
// emb_table[f32], W1[f32], b1[f32], W2[f32], b2[f32]. Output: f32 [8192,1].
// Workspace: pooled [8192,128] f32 = 4 MB at d_ws.
// ---------------------------------------------------------------------------
extern "C" void kernel_launch(void* const* d_in, const int* in_sizes, int n_in,
                              void* d_out, int out_size, void* d_ws, size_t ws_size,
                              hipStream_t stream) {
  const int*   tokens  = (const int*)d_in[0];
  const int*   lengths = (const int*)d_in[1];
  const float* emb     = (const float*)d_in[2];
  const float* W1      = (const float*)d_in[3];
  const float* b1      = (const float*)d_in[4];
  const float* W2      = (const float*)d_in[5];
  const float* b2      = (const float*)d_in[6];
  float* out    = (float*)d_out;
  float* pooled = (float*)d_ws;  // 8192*128*4 = 4 MB

  // 1 wave per batch row; 256 threads = 8 waves per block.
  embbag_mean_kernel<<<BATCH / 8, 256, 0, stream>>>(tokens, lengths, emb, pooled);

  // 512 M-tiles (8192/16), 1 wave each, 8 waves per block -> 64 blocks.
  mlp_wmma_kernel<<<(BATCH / 16) / 8, 256, 0, stream>>>(pooled, W1, b1, W2, b2, out);
}